// Wireless_Autoencoder_87608742903914
// MI455X (gfx1250) — compile-verified
//
#include <hip/hip_runtime.h>
#include <hip/hip_bf16.h>

typedef float v2f __attribute__((ext_vector_type(2)));
typedef float v8f __attribute__((ext_vector_type(8)));

#define DEV __device__ __forceinline__

// ---------------- fast activations (gfx1250 trans ops) ----------------
DEV float fast_tanh(float x) {
#if __has_builtin(__builtin_amdgcn_tanhf)
  return __builtin_amdgcn_tanhf(x);
#else
  float r;
  // v_tanh_f32 is a TRANS32 op; keep a v_nop after it so the result is not
  // consumed in the co-execution shadow (ISA 15.5 trans hazard rule).
  asm("v_tanh_f32 %0, %1\n\tv_nop" : "=v"(r) : "v"(x));
  return r;
#endif
}

DEV float fast_elu(float x) { return x > 0.0f ? x : (__expf(x) - 1.0f); }

// ---------------- WMMA f32 16x16x4 dense layers ----------------
// Layout facts (ISA 7.12.2, f32):
//   A[m][k] : lane (k/2)*16 + m , vgpr k%2      (16x4 tile, K-chunk t => k = 4t+2*half+{0,1})
//   B[k][n] : lane (k/2)*16 + n , vgpr k%2      (4x16 tile)
//   D[m][n] : lane (m/8)*16 + n , vgpr m%8
// LDS staging is sample-major: stage[sample][feature], feature-stride ST (=F+4, 16B aligned).
// One wave owns 32 samples = two 16-sample tiles; all LDS traffic is same-wave (in-order, no barrier).

template<int K, int M, int MR, int ACT, int STIN, int STOUT>
DEV void dense_layer(const float* __restrict__ W, const float* __restrict__ bias, float* st) {
  const int lane = threadIdx.x & 31;
  const int nl   = lane & 15;
  const int half = lane >> 4;
  const int off  = half * 2;
  constexpr int NM = M / 16;
  constexpr int NK = K / 4;
  v8f d[NM][2];
  // ---- compute phase: read all B operands before any store (in-place safe) ----
#pragma unroll
  for (int mt = 0; mt < NM; ++mt) {
    const int  mrow  = mt * 16 + nl;
    const bool valid = (MR == M) || (mrow < MR);
    const float am   = valid ? 1.0f : 0.0f;
    const float* wr  = W + (valid ? mrow : 0) * K;
#pragma unroll
    for (int T = 0; T < 2; ++T) {
      v8f acc = {};
#pragma unroll
      for (int t = 0; t < NK; ++t) {
        v2f a = *(const v2f*)(wr + 4 * t + off);
        a *= am;                                   // zero-pad rows beyond MR (no EXEC divergence)
        v2f b = *(const v2f*)(st + (T * 16 + nl) * STIN + 4 * t + off);
        acc = __builtin_amdgcn_wmma_f32_16x16x4_f32(false, a, false, b, (short)0, acc, false, false);
      }
      d[mt][T] = acc;
    }
  }
  // ---- bias + activation + store phase ----
#pragma unroll
  for (int mt = 0; mt < NM; ++mt) {
    const int mb = mt * 16 + half * 8;
    float bv[8];
#pragma unroll
    for (int r = 0; r < 8; ++r) {
      const int  m = mb + r;
      const bool v = (MR == M) || (m < MR);
      bv[r] = bias[v ? m : 0] * (v ? 1.0f : 0.0f);
    }
#pragma unroll
    for (int T = 0; T < 2; ++T) {
      float* o = st + (T * 16 + nl) * STOUT + mb;
#pragma unroll
      for (int r = 0; r < 8; ++r) {
        float vv = d[mt][T][r] + bv[r];
        if (ACT == 1)      vv = fast_elu(vv);
        else if (ACT == 2) vv = fast_tanh(vv);
        o[r] = vv;
      }
    }
  }
}

// 16-output dense, result kept in D-register layout (for direct global store / reductions)
template<int K, int STIN>
DEV void dense16_reg(const float* __restrict__ W, const float* __restrict__ bias,
                     const float* st, v8f& o0, v8f& o1) {
  const int lane = threadIdx.x & 31;
  const int nl   = lane & 15;
  const int half = lane >> 4;
  const int off  = half * 2;
  constexpr int NK = K / 4;
  const float* wr = W + nl * K;
  v8f acc0 = {}, acc1 = {};
#pragma unroll
  for (int t = 0; t < NK; ++t) {
    v2f a  = *(const v2f*)(wr + 4 * t + off);
    v2f b0 = *(const v2f*)(st + nl * STIN + 4 * t + off);
    v2f b1 = *(const v2f*)(st + (16 + nl) * STIN + 4 * t + off);
    acc0 = __builtin_amdgcn_wmma_f32_16x16x4_f32(false, a, false, b0, (short)0, acc0, false, false);
    acc1 = __builtin_amdgcn_wmma_f32_16x16x4_f32(false, a, false, b1, (short)0, acc1, false, false);
  }
#pragma unroll
  for (int r = 0; r < 8; ++r) {
    const float bv = bias[half * 8 + r];
    acc0[r] += bv; acc1[r] += bv;
  }
  o0 = acc0; o1 = acc1;
}

// ---------------- per-lane conv stack: 1ch x16 -> 8ch x4 (flattened 32) ----------------
DEV void conv_chain(const float h[16],
                    const float* __restrict__ w1, const float* __restrict__ b1,
                    const float* __restrict__ w2, const float* __restrict__ b2,
                    const float* __restrict__ w3, const float* __restrict__ b3,
                    const float* __restrict__ w4, const float* __restrict__ b4,
                    float out[32]) {
  float c2[8][6];
#pragma unroll
  for (int i = 0; i < 6; ++i) {
    float t1[8][4];                               // recompute conv1 block (caps registers)
#pragma unroll
    for (int ic = 0; ic < 8; ++ic) {
      const float wa = w1[2 * ic], wb = w1[2 * ic + 1], bb = b1[ic];
#pragma unroll
      for (int j = 0; j < 4; ++j) {
        const int p = 2 * i + j;
        t1[ic][j] = fast_tanh(bb + wa * h[p] + wb * h[p + 1]);
      }
    }
#pragma unroll
    for (int oc = 0; oc < 8; ++oc) {
      float acc = b2[oc];
#pragma unroll
      for (int ic = 0; ic < 8; ++ic)
#pragma unroll
        for (int j = 0; j < 4; ++j)
          acc += w2[oc * 32 + ic * 4 + j] * t1[ic][j];
      c2[oc][i] = fast_tanh(acc);
    }
  }
  float c3[8][5];
#pragma unroll
  for (int i = 0; i < 5; ++i)
#pragma unroll
    for (int oc = 0; oc < 8; ++oc) {
      float acc = b3[oc];
#pragma unroll
      for (int ic = 0; ic < 8; ++ic)
        acc += w3[oc * 16 + ic * 2] * c2[ic][i] + w3[oc * 16 + ic * 2 + 1] * c2[ic][i + 1];
      c3[oc][i] = fast_tanh(acc);
    }
#pragma unroll
  for (int i = 0; i < 4; ++i)
#pragma unroll
    for (int oc = 0; oc < 8; ++oc) {
      float acc = b4[oc];
#pragma unroll
      for (int ic = 0; ic < 8; ++ic)
        acc += w4[oc * 16 + ic * 2] * c3[ic][i] + w4[oc * 16 + ic * 2 + 1] * c3[ic][i + 1];
      out[oc * 4 + i] = fast_tanh(acc);
    }
}

#define WAVES 4   // 128 threads/block = 4 waves; 32 samples per wave

// ---------------- kernel 0: zero global BN accumulators ----------------
__global__ void zero_kernel(float* p) {
  if (threadIdx.x < 32) p[threadIdx.x] = 0.0f;
}

// ---------------- kernel 1: encoder up to ew3; h -> scratch; BN partial sums ----------------
__global__ void __launch_bounds__(128) enc_kernel(
    const float* __restrict__ x,
    const float* __restrict__ ew1, const float* __restrict__ eb1,
    const float* __restrict__ ew2, const float* __restrict__ eb2,
    const float* __restrict__ c1w, const float* __restrict__ c1b,
    const float* __restrict__ c2w, const float* __restrict__ c2b,
    const float* __restrict__ c3w, const float* __restrict__ c3b,
    const float* __restrict__ c4w, const float* __restrict__ c4b,
    const float* __restrict__ ew3, const float* __restrict__ eb3,
    float* __restrict__ hbuf, float* __restrict__ gsum, float* __restrict__ gsq) {
  __shared__ __align__(16) float stage[WAVES][32 * 36];
  __shared__ float bsum[16], bsq[16];
  const int tid = threadIdx.x;
  if (tid < 16) bsum[tid] = 0.0f;
  else if (tid < 32) bsq[tid - 16] = 0.0f;
  __syncthreads();

  const int wave = tid >> 5, lane = tid & 31;
  const int nl = lane & 15, half = lane >> 4;
  float* st = stage[wave];
  const long g0 = (long)blockIdx.x * 128 + wave * 32;   // first sample of this wave's group
  const long s  = g0 + lane;                            // this lane's sample

  // x -> stage (ST=20)
  {
    const float* xp = x + s * 16;
#pragma unroll
    for (int i = 0; i < 16; ++i) st[lane * 20 + i] = xp[i];
  }
  dense_layer<16, 16, 16, 1, 20, 20>(ew1, eb1, st);     // elu
  dense_layer<16, 16, 16, 1, 20, 20>(ew2, eb2, st);     // elu

  float h[16];
#pragma unroll
  for (int i = 0; i < 16; ++i) h[i] = st[lane * 20 + i];
  float e4[32];
  conv_chain(h, c1w, c1b, c2w, c2b, c3w, c3b, c4w, c4b, e4);
#pragma unroll
  for (int i = 0; i < 32; ++i) st[lane * 36 + i] = e4[i];

  v8f o0, o1;
  dense16_reg<32, 36>(ew3, eb3, st, o0, o1);            // h = e4 @ ew3.T + eb3 (no act)

  // write h tiles directly from D layout: hbuf[(sample)*16 + m]
  {
    float* p0 = hbuf + (g0 + nl) * 16 + half * 8;
    float* p1 = hbuf + (g0 + 16 + nl) * 16 + half * 8;
#pragma unroll
    for (int r = 0; r < 8; ++r) { p0[r] = o0[r]; p1[r] = o1[r]; }
  }

  // BN partial sums: reduce over samples (lanes within each 16-half)
  float sv[8], qv[8];
#pragma unroll
  for (int r = 0; r < 8; ++r) {
    sv[r] = o0[r] + o1[r];
    qv[r] = o0[r] * o0[r] + o1[r] * o1[r];
  }
#pragma unroll
  for (int r = 0; r < 8; ++r) {
    sv[r] += __shfl_xor(sv[r], 1); sv[r] += __shfl_xor(sv[r], 2);
    sv[r] += __shfl_xor(sv[r], 4); sv[r] += __shfl_xor(sv[r], 8);
    qv[r] += __shfl_xor(qv[r], 1); qv[r] += __shfl_xor(qv[r], 2);
    qv[r] += __shfl_xor(qv[r], 4); qv[r] += __shfl_xor(qv[r], 8);
  }
  if (nl == 0) {
#pragma unroll
    for (int r = 0; r < 8; ++r) {
      atomicAdd(&bsum[half * 8 + r], sv[r]);
      atomicAdd(&bsq [half * 8 + r], qv[r]);
    }
  }
  __syncthreads();
  if (tid < 16)       atomicAdd(&gsum[tid], bsum[tid]);
  else if (tid < 32)  atomicAdd(&gsq[tid - 16], bsq[tid - 16]);
}

// ---------------- kernel 2: fold BN stats into scale/shift ----------------
__global__ void bn_kernel(const float* __restrict__ acc,
                          const float* __restrict__ bng, const float* __restrict__ bnb,
                          float* __restrict__ sc_sh, float invB) {
  const int m = threadIdx.x;
  if (m < 16) {
    const float mu  = acc[m] * invB;
    const float var = acc[16 + m] * invB - mu * mu;
    const float sc  = bng[m] * rsqrtf(var + 1e-5f);
    sc_sh[m]      = sc;
    sc_sh[16 + m] = bnb[m] - mu * sc;
  }
}

// ---------------- kernel 3: BN apply + channel + p-net + equalizer + decoder ----------------
__global__ void __launch_bounds__(128) dec_kernel(
    const float* __restrict__ hbuf, const float* __restrict__ noise, const float* __restrict__ fading,
    const float* __restrict__ scale, const float* __restrict__ shift,
    const float* __restrict__ pw1, const float* __restrict__ pb1,
    const float* __restrict__ pw2, const float* __restrict__ pb2,
    const float* __restrict__ pw3, const float* __restrict__ pb3,
    const float* __restrict__ pw4, const float* __restrict__ pb4,
    const float* __restrict__ dw1, const float* __restrict__ db1,
    const float* __restrict__ c1w, const float* __restrict__ c1b,
    const float* __restrict__ c2w, const float* __restrict__ c2b,
    const float* __restrict__ c3w, const float* __restrict__ c3b,
    const float* __restrict__ c4w, const float* __restrict__ c4b,
    const float* __restrict__ dw2, const float* __restrict__ db2,
    const float* __restrict__ dw3, const float* __restrict__ db3,
    const float* __restrict__ dw4, const float* __restrict__ db4,
    float* __restrict__ out) {
  __shared__ __align__(16) float stage[WAVES][32 * 68];
  const int tid = threadIdx.x, wave = tid >> 5, lane = tid & 31;
  const int nl = lane & 15, half = lane >> 4;
  float* st = stage[wave];
  const long g0 = (long)blockIdx.x * 128 + wave * 32;
  const long s  = g0 + lane;

  // normalize
  float enc[16];
  {
    const float* hp = hbuf + s * 16;
#pragma unroll
    for (int i = 0; i < 16; ++i) enc[i] = hp[i] * scale[i] + shift[i];
  }
  // fading taps reversed: ff[t] = fading[2-t]
  float fr[3], fi[3];
  {
    const float* fp = fading + s * 6;
#pragma unroll
    for (int t = 0; t < 3; ++t) { fr[t] = fp[(2 - t) * 2]; fi[t] = fp[(2 - t) * 2 + 1]; }
  }
  // complex conv channel + noise -> c[16]
  float c[16];
  {
    const float* np = noise + s * 16;
#pragma unroll
    for (int l = 0; l < 8; ++l) {
      float yr = 0.0f, yi = 0.0f;
#pragma unroll
      for (int t = 0; t < 3; ++t) {
        const int p = l + t - 2;                    // constant-folds per (l,t)
        if (p >= 0 && p < 8) {
          const float wr = enc[2 * p], wi = enc[2 * p + 1];
          yr += wr * fr[t] - wi * fi[t];
          yi += wr * fi[t] + wi * fr[t];
        }
      }
      c[2 * l]     = yr + np[2 * l];
      c[2 * l + 1] = yi + np[2 * l + 1];
    }
  }
  // p-net: 16 -> 32(elu) -> 64(tanh) -> 8(tanh, padded to 16) -> 2
#pragma unroll
  for (int i = 0; i < 16; ++i) st[lane * 20 + i] = c[i];
  dense_layer<16, 32, 32, 1, 20, 36>(pw1, pb1, st);
  dense_layer<32, 64, 64, 2, 36, 68>(pw2, pb2, st);
  dense_layer<64, 16,  8, 2, 68, 20>(pw3, pb3, st);   // rows 8..15 zero-padded
  float p3[8];
#pragma unroll
  for (int i = 0; i < 8; ++i) p3[i] = st[lane * 20 + i];
  float hr = pb4[0], hi = pb4[1];
#pragma unroll
  for (int k = 0; k < 8; ++k) { hr += pw4[k] * p3[k]; hi += pw4[8 + k] * p3[k]; }
  // equalizer
  const float den = hr * hr + hi * hi;
  float t16[16];
#pragma unroll
  for (int l = 0; l < 8; ++l) {
    const float cr = c[2 * l], ci = c[2 * l + 1];
    t16[2 * l]     = (cr * hr + ci * hi) / den;
    t16[2 * l + 1] = (ci * hr - cr * hi) / den;
  }
  // decoder
#pragma unroll
  for (int i = 0; i < 16; ++i) st[lane * 20 + i] = t16[i];
  dense_layer<16, 16, 16, 2, 20, 20>(dw1, db1, st);
  float d1[16];
#pragma unroll
  for (int i = 0; i < 16; ++i) d1[i] = st[lane * 20 + i];
  float e4[32];
  conv_chain(d1, c1w, c1b, c2w, c2b, c3w, c3b, c4w, c4b, e4);
#pragma unroll
  for (int i = 0; i < 32; ++i) st[lane * 36 + i] = e4[i];
  dense_layer<32, 16, 16, 2, 36, 20>(dw2, db2, st);
  dense_layer<16, 16, 16, 2, 20, 20>(dw3, db3, st);
  v8f o0, o1;
  dense16_reg<16, 20>(dw4, db4, st, o0, o1);
  // final store straight from D layout
  {
    float* p0 = out + (g0 + nl) * 16 + half * 8;
    float* p1 = out + (g0 + 16 + nl) * 16 + half * 8;
#pragma unroll
    for (int r = 0; r < 8; ++r) { p0[r] = o0[r]; p1[r] = o1[r]; }
  }
}

extern "C" void kernel_launch(void* const* d_in, const int* in_sizes, int n_in,
                              void* d_out, int out_size, void* d_ws, size_t ws_size,
                              hipStream_t stream) {
  (void)n_in; (void)out_size; (void)ws_size;
  const float* x     = (const float*)d_in[0];
  const float* noise = (const float*)d_in[1];
  const float* fad   = (const float*)d_in[2];
  const float* ew1 = (const float*)d_in[3];  const float* eb1 = (const float*)d_in[4];
  const float* ew2 = (const float*)d_in[5];  const float* eb2 = (const float*)d_in[6];
  const float* ec1w = (const float*)d_in[7];  const float* ec1b = (const float*)d_in[8];
  const float* ec2w = (const float*)d_in[9];  const float* ec2b = (const float*)d_in[10];
  const float* ec3w = (const float*)d_in[11]; const float* ec3b = (const float*)d_in[12];
  const float* ec4w = (const float*)d_in[13]; const float* ec4b = (const float*)d_in[14];
  const float* ew3 = (const float*)d_in[15]; const float* eb3 = (const float*)d_in[16];
  const float* bng = (const float*)d_in[17]; const float* bnb = (const float*)d_in[18];
  const float* pw1 = (const float*)d_in[19]; const float* pb1 = (const float*)d_in[20];
  const float* pw2 = (const float*)d_in[21]; const float* pb2 = (const float*)d_in[22];
  const float* pw3 = (const float*)d_in[23]; const float* pb3 = (const float*)d_in[24];
  const float* pw4 = (const float*)d_in[25]; const float* pb4 = (const float*)d_in[26];
  const float* dw1 = (const float*)d_in[27]; const float* db1 = (const float*)d_in[28];
  const float* dc1w = (const float*)d_in[29]; const float* dc1b = (const float*)d_in[30];
  const float* dc2w = (const float*)d_in[31]; const float* dc2b = (const float*)d_in[32];
  const float* dc3w = (const float*)d_in[33]; const float* dc3b = (const float*)d_in[34];
  const float* dc4w = (const float*)d_in[35]; const float* dc4b = (const float*)d_in[36];
  const float* dw2 = (const float*)d_in[37]; const float* db2 = (const float*)d_in[38];
  const float* dw3 = (const float*)d_in[39]; const float* db3 = (const float*)d_in[40];
  const float* dw4 = (const float*)d_in[41]; const float* db4 = (const float*)d_in[42];

  float* wsf = (float*)d_ws;
  // ws layout: [0..15] sum, [16..31] sumsq, [32..47] scale, [48..63] shift, [64..) h scratch (B*16)
  const int B = in_sizes[0] / 16;
  const int nblk = B / 128;   // B = 131072 -> 1024 blocks of 4 waves x 32 samples

  zero_kernel<<<1, 64, 0, stream>>>(wsf);
  enc_kernel<<<nblk, 128, 0, stream>>>(x, ew1, eb1, ew2, eb2,
                                       ec1w, ec1b, ec2w, ec2b, ec3w, ec3b, ec4w, ec4b,
                                       ew3, eb3, wsf + 64, wsf, wsf + 16);
  bn_kernel<<<1, 16, 0, stream>>>(wsf, bng, bnb, wsf + 32, 1.0f / (float)B);
  dec_kernel<<<nblk, 128, 0, stream>>>(wsf + 64, noise, fad, wsf + 32, wsf + 48,
                                       pw1, pb1, pw2, pb2, pw3, pb3, pw4, pb4,
                                       dw1, db1,
                                       dc1w, dc1b, dc2w, dc2b, dc3w, dc3b, dc4w, dc4b,
                                       dw2, db2, dw3, db3, dw4, db4,
                                       (float*)d_out);
}